// MultiHeadAttention_30983894073461
// MI455X (gfx1250) — compile-verified
//
#include <hip/hip_runtime.h>
#include <stdint.h>

// -------------------------------------------------------------------------
// MultiHeadAttention for MI455X (gfx1250, wave32, WMMA).
// All matmuls use V_WMMA_F32_16X16X4_F32 (fp32 in / fp32 acc): ~52 GFLOP vs a
// working set (~110MB) that fits in the 192MB L2 -> bandwidth-bound, so full
// fp32 matrix precision is free.  LDS staging uses CDNA5's async memory->LDS
// path (GLOBAL_LOAD_ASYNC_TO_LDS_B128, ASYNCcnt) with double buffering so the
// copy of block k+1 overlaps the WMMAs on block k.
//
// d_ws layout (floats):  Q | K | V  each [B,H,T,HD] = 4M elems, then C [B,T,D]
// -> needs 4 * 16MB = 64MB of workspace.
// -------------------------------------------------------------------------

typedef __attribute__((ext_vector_type(2))) float v2f;
typedef __attribute__((ext_vector_type(8))) float v8f;

#define DIMD  1024
#define HEADS 16
#define HDIM  64
#define SEQT  2048
#define BATCH 2

__device__ __forceinline__ v8f wmma_f32(v2f a, v2f b, v8f c) {
  // (neg_a, A, neg_b, B, c_mod, C, reuse_a, reuse_b)
  return __builtin_amdgcn_wmma_f32_16x16x4_f32(false, a, false, b, (short)0, c,
                                               false, false);
}

// Async 16-byte copy global -> LDS (tracked by ASYNCcnt, no VGPR round-trip).
// LDS address = low 32 bits of the generic pointer (flat aperture rule).
__device__ __forceinline__ void async_copy16(void* lds_ptr, const float* gp) {
  uint32_t la = (uint32_t)(uintptr_t)lds_ptr;
  uint64_t ga = (uint64_t)(uintptr_t)gp;
  asm volatile("global_load_async_to_lds_b128 %0, %1, off"
               :
               : "v"(la), "v"(ga)
               : "memory");
}
__device__ __forceinline__ void wait_async() {
  asm volatile("s_wait_asynccnt 0" ::: "memory");
}

// ============================ GEMM ======================================
// out[M,N] = A[M,K] @ Bw[K,N]  (all row-major).
// mode 0: out row-major [M,N], += bias[col] if bias != nullptr.
// mode 1: scatter to [B,H,T,HD]: row = b*T+t, col = h*HD+hd.
// Block: 256 threads = 8 waves; block tile 128(M) x 64(N); wave tile 32x32.
// K-blocked x16 through double-buffered LDS filled by async copies.
#define GTM 128
#define GTN 64
#define GKB 16
#define APAD 4  // pad to 20 floats: 16B-aligned rows + conflict-free gathers

__global__ __launch_bounds__(256) void gemm_f32_wmma_kernel(
    const float* __restrict__ A, const float* __restrict__ Bw,
    float* __restrict__ out, const float* __restrict__ bias, int M, int N,
    int K, int mode) {
  __shared__ float Asm[2][GTM][GKB + APAD];
  __shared__ float Bsm[2][GKB][GTN];

  const int tid = threadIdx.x;
  const int wave = tid >> 5;
  const int lane = tid & 31;
  const int l = lane & 15;
  const int hh = lane >> 4;  // half-wave select (K split / row+8 split)
  const int wm = (wave >> 1) * 32;
  const int wn = (wave & 1) * 32;
  const int mbase = blockIdx.y * GTM;
  const int nbase = blockIdx.x * GTN;

  // staging coordinates (fixed per thread)
  const int ar0 = tid >> 2, ac0 = (tid & 3) * 4;             // A chunk 0
  const int ar1 = (tid + 256) >> 2, ac1 = ((tid + 256) & 3) * 4;  // A chunk 1
  const int br = tid >> 4, bc = (tid & 15) * 4;              // B chunk

  auto stage = [&](int buf, int k0) {
    async_copy16(&Asm[buf][ar0][ac0], A + (size_t)(mbase + ar0) * K + k0 + ac0);
    async_copy16(&Asm[buf][ar1][ac1], A + (size_t)(mbase + ar1) * K + k0 + ac1);
    async_copy16(&Bsm[buf][br][bc], Bw + (size_t)(k0 + br) * N + nbase + bc);
  };

  v8f acc[2][2] = {};

  stage(0, 0);
  wait_async();
  __syncthreads();

  int cur = 0;
  for (int k0 = 0; k0 < K; k0 += GKB) {
    if (k0 + GKB < K) stage(cur ^ 1, k0 + GKB);  // overlap with compute

#pragma unroll
    for (int kk = 0; kk < GKB; kk += 4) {
      // A fragment (16x4 fp32): v0 = K+2h, v1 = K+2h+1, row = lane%16
      v2f a0 = {Asm[cur][wm + l][kk + 2 * hh],
                Asm[cur][wm + l][kk + 2 * hh + 1]};
      v2f a1 = {Asm[cur][wm + 16 + l][kk + 2 * hh],
                Asm[cur][wm + 16 + l][kk + 2 * hh + 1]};
      // B fragment (4x16 fp32): v0 row K+2h, v1 row K+2h+1, col = lane%16
      v2f b0 = {Bsm[cur][kk + 2 * hh][wn + l],
                Bsm[cur][kk + 2 * hh + 1][wn + l]};
      v2f b1 = {Bsm[cur][kk + 2 * hh][wn + 16 + l],
                Bsm[cur][kk + 2 * hh + 1][wn + 16 + l]};
      acc[0][0] = wmma_f32(a0, b0, acc[0][0]);
      acc[0][1] = wmma_f32(a0, b1, acc[0][1]);
      acc[1][0] = wmma_f32(a1, b0, acc[1][0]);
      acc[1][1] = wmma_f32(a1, b1, acc[1][1]);
    }

    wait_async();     // prefetched block fully landed in LDS
    __syncthreads();  // and everyone is done reading the current one
    cur ^= 1;
  }

  // ---- store: C/D layout: VGPR r -> row r (lanes 0-15) / r+8 (lanes 16-31)
#pragma unroll
  for (int i = 0; i < 2; ++i)
#pragma unroll
    for (int j = 0; j < 2; ++j)
#pragma unroll
      for (int r = 0; r < 8; ++r) {
        const int row = mbase + wm + 16 * i + r + 8 * hh;
        const int col = nbase + wn + 16 * j + l;
        float v = acc[i][j][r];
        if (mode == 0) {
          if (bias) v += bias[col];
          out[(size_t)row * N + col] = v;
        } else {
          const int b = row / SEQT;
          const int t = row - b * SEQT;
          const int h = col >> 6;
          const int hd = col & 63;
          out[(((size_t)b * HEADS + h) * SEQT + t) * HDIM + hd] = v;
        }
      }
}

// ======================= Flash attention ================================
// One 128-thread block (4 waves) per (b*h, 64-query tile).
// Wave w owns query rows [q0+16w, q0+16w+15]; online softmax state in regs.
// K/V blocks double-buffered in LDS via async copies.
#define VPAD 4  // 68-float rows: 16B-aligned chunks + conflict-free gathers

__global__ __launch_bounds__(128) void attn_f32_wmma_kernel(
    const float* __restrict__ Q, const float* __restrict__ K,
    const float* __restrict__ V, float* __restrict__ Cout) {
  const int bh = blockIdx.y;
  const int b = bh / HEADS;
  const int h = bh - b * HEADS;
  const int q0 = blockIdx.x * 64;
  const float* Qh = Q + (size_t)bh * SEQT * HDIM;
  const float* Kh = K + (size_t)bh * SEQT * HDIM;
  const float* Vh = V + (size_t)bh * SEQT * HDIM;

  __shared__ float Qs[64][HDIM + VPAD];
  __shared__ float Ks[2][64][HDIM + VPAD];
  __shared__ float Vs[2][64][HDIM + VPAD];
  __shared__ float Ps[4][16][17];  // per-wave P tile restage (C-layout -> A)

  const int tid = threadIdx.x;
  const int wave = tid >> 5;
  const int lane = tid & 31;
  const int l = lane & 15;
  const int hh = lane >> 4;
  const int qbase = q0 + wave * 16;

  // ---- async staging: 64x64 tile = 1024 float4 chunks / 128 threads ----
  auto stageKV = [&](int buf, int kb0) {
#pragma unroll
    for (int it = 0; it < 8; ++it) {
      const int j = tid + 128 * it;
      const int row = j >> 4;
      const int c4 = (j & 15) * 4;
      async_copy16(&Ks[buf][row][c4], Kh + (size_t)(kb0 + row) * HDIM + c4);
      async_copy16(&Vs[buf][row][c4], Vh + (size_t)(kb0 + row) * HDIM + c4);
    }
  };

  // stage Q tile + first K/V block
#pragma unroll
  for (int it = 0; it < 8; ++it) {
    const int j = tid + 128 * it;
    const int row = j >> 4;
    const int c4 = (j & 15) * 4;
    async_copy16(&Qs[row][c4], Qh + (size_t)(q0 + row) * HDIM + c4);
  }
  stageKV(0, 0);

  v8f o[4] = {};  // 16 rows x 64 cols of output, 4 WMMA C-tiles
  float m_i[8], l_i[8];
#pragma unroll
  for (int r = 0; r < 8; ++r) {
    m_i[r] = -1e30f;
    l_i[r] = 0.0f;
  }

  wait_async();
  __syncthreads();

  const int nblk = blockIdx.x + 1;  // causal: keys only up to q0+63
  int cur = 0;
  for (int blk = 0; blk < nblk; ++blk) {
    const int kb0 = blk * 64;
    if (blk + 1 < nblk) stageKV(cur ^ 1, kb0 + 64);  // overlap with compute

#pragma unroll 1
    for (int s = 0; s < 4; ++s) {
      const int kk0 = kb0 + s * 16;  // global key base of this 16-key tile
      if (kk0 > qbase + 15) continue;  // wave-uniform skip: EXEC stays all-1s
      const int kl = s * 16;

      // ---- S(16x16) = Q(16x64) @ K^T : 16 fp32 WMMAs over the head dim ----
      v8f sacc = {};
#pragma unroll
      for (int kd = 0; kd < HDIM; kd += 4) {
        v2f a = {Qs[wave * 16 + l][kd + 2 * hh],
                 Qs[wave * 16 + l][kd + 2 * hh + 1]};
        v2f bb = {Ks[cur][kl + l][kd + 2 * hh],
                  Ks[cur][kl + l][kd + 2 * hh + 1]};
        sacc = wmma_f32(a, bb, sacc);
      }

      // ---- causal mask + 1/sqrt(HD) scale ----
#pragma unroll
      for (int r = 0; r < 8; ++r) {
        const int qrow = qbase + r + 8 * hh;
        const int kcol = kk0 + l;
        float vsc = sacc[r] * 0.125f;  // 1/sqrt(64)
        if (kcol > qrow) vsc = -1e30f;
        sacc[r] = vsc;
      }

      // ---- online softmax: row max over the 16 lanes of each half-group ---
      float bm[8];
#pragma unroll
      for (int r = 0; r < 8; ++r) bm[r] = sacc[r];
#pragma unroll
      for (int off = 1; off < 16; off <<= 1)
#pragma unroll
        for (int r = 0; r < 8; ++r)
          bm[r] = fmaxf(bm[r], __shfl_xor(bm[r], off, 32));

      float p[8], alpha[8];
#pragma unroll
      for (int r = 0; r < 8; ++r) {
        const float mnew = fmaxf(m_i[r], bm[r]);
        alpha[r] = __expf(m_i[r] - mnew);
        m_i[r] = mnew;
        p[r] = __expf(sacc[r] - mnew);
      }
      float rs[8];
#pragma unroll
      for (int r = 0; r < 8; ++r) rs[r] = p[r];
#pragma unroll
      for (int off = 1; off < 16; off <<= 1)
#pragma unroll
        for (int r = 0; r < 8; ++r) rs[r] += __shfl_xor(rs[r], off, 32);
#pragma unroll
      for (int r = 0; r < 8; ++r) {
        l_i[r] = l_i[r] * alpha[r] + rs[r];
#pragma unroll
        for (int t = 0; t < 4; ++t) o[t][r] *= alpha[r];
        Ps[wave][r + 8 * hh][l] = p[r];  // restage P: C-layout -> A-layout
      }

      // ---- O(16x64) += P(16x16) @ V(16x64): 16 fp32 WMMAs ----
#pragma unroll
      for (int kk = 0; kk < 16; kk += 4) {
        v2f a = {Ps[wave][l][kk + 2 * hh], Ps[wave][l][kk + 2 * hh + 1]};
#pragma unroll
        for (int t = 0; t < 4; ++t) {
          v2f bb = {Vs[cur][kl + kk + 2 * hh][t * 16 + l],
                    Vs[cur][kl + kk + 2 * hh + 1][t * 16 + l]};
          o[t] = wmma_f32(a, bb, o[t]);
        }
      }
    }

    wait_async();     // next K/V block fully landed
    __syncthreads();  // everyone done reading the current one
    cur ^= 1;
  }

  // ---- normalize and write C in [B,T,D] row-major ----
#pragma unroll
  for (int r = 0; r < 8; ++r) {
    const int qrow = qbase + r + 8 * hh;
    const float inv = 1.0f / l_i[r];
#pragma unroll
    for (int t = 0; t < 4; ++t) {
      const int col = t * 16 + l;
      Cout[(size_t)(b * SEQT + qrow) * DIMD + h * HDIM + col] = o[t][r] * inv;
    }
  }
}

// ============================ launcher ==================================
extern "C" void kernel_launch(void* const* d_in, const int* in_sizes, int n_in,
                              void* d_out, int out_size, void* d_ws,
                              size_t ws_size, hipStream_t stream) {
  const float* X = (const float*)d_in[0];
  const float* Wq = (const float*)d_in[1];
  const float* Wk = (const float*)d_in[2];
  const float* Wv = (const float*)d_in[3];
  const float* Wo = (const float*)d_in[4];
  const float* bo = (const float*)d_in[5];
  float* out = (float*)d_out;

  const size_t perBuf = (size_t)BATCH * HEADS * SEQT * HDIM;  // 4M floats
  float* Qb = (float*)d_ws;
  float* Kb = Qb + perBuf;
  float* Vb = Kb + perBuf;
  float* Cb = Vb + perBuf;  // [B*T, D] row-major

  const int M = BATCH * SEQT;  // 4096
  const int N = DIMD;          // 1024
  const int K = DIMD;          // 1024

  dim3 gblk(256);
  dim3 ggrid(N / GTN, M / GTM);  // (16, 32)
  gemm_f32_wmma_kernel<<<ggrid, gblk, 0, stream>>>(X, Wq, Qb, nullptr, M, N, K, 1);
  gemm_f32_wmma_kernel<<<ggrid, gblk, 0, stream>>>(X, Wk, Kb, nullptr, M, N, K, 1);
  gemm_f32_wmma_kernel<<<ggrid, gblk, 0, stream>>>(X, Wv, Vb, nullptr, M, N, K, 1);

  dim3 ablk(128);
  dim3 agrid(SEQT / 64, BATCH * HEADS);  // (32, 32)
  attn_f32_wmma_kernel<<<agrid, ablk, 0, stream>>>(Qb, Kb, Vb, Cb);

  gemm_f32_wmma_kernel<<<ggrid, gblk, 0, stream>>>(Cb, Wo, out, bo, M, N, K, 0);
}